// ExiLU_14147622273111
// MI455X (gfx1250) — compile-verified
//
#include <hip/hip_runtime.h>
#include <stdint.h>

// ---------------------------------------------------------------------------
// Element-wise select (tautological for finite x) over a 4096x32768 fp32
// tensor. Pure HBM-streaming problem: 1 GiB total traffic, ~46 us floor at
// 23.3 TB/s. Strategy: CDNA5 async global->LDS pipeline (ASYNCcnt, double
// buffered) + b128 per-lane transactions + non-temporal stores.
// ---------------------------------------------------------------------------

typedef __attribute__((ext_vector_type(4))) float v4f;

#define BLOCK 256
#define GRID  8192

__device__ __forceinline__ float sel_one(float x) {
    // Faithful to reference: (x >= MIN_ITEMS) | (x <= MAX_ITEMS) ? x : 0
    bool c = (x >= 1001.0f) || (x <= 447113.0f);
    return c ? x : 0.0f;
}

// Bulk kernel: every thread performs exactly `iters` iterations (uniform,
// non-divergent). Tile t+1 is async-loaded into the alternate LDS buffer
// while tile t is consumed (software pipeline depth 2 per wave; occupancy
// supplies the rest of the MLP).
__global__ void __launch_bounds__(BLOCK) copy_select_async(
    const v4f* __restrict__ in, v4f* __restrict__ out,
    int iters, long long stride /* in float4 units == total threads */)
{
    __shared__ v4f smem[2][BLOCK];

    const unsigned tid = threadIdx.x;
    long long idx = (long long)blockIdx.x * BLOCK + tid;

    // Low 32 bits of the flat pointer == wave-relative LDS byte address
    // (LDS aperture occupies the high 32 bits per the gfx1250 aperture check).
    unsigned lds0 = (unsigned)(uintptr_t)&smem[0][tid];
    unsigned lds1 = (unsigned)(uintptr_t)&smem[1][tid];

    // Prologue: async-load first tile into buffer 0.
    {
        unsigned long long g = (unsigned long long)(uintptr_t)(in + idx);
        asm volatile("global_load_async_to_lds_b128 %0, %1, off"
                     :: "v"(lds0), "v"(g) : "memory");
    }

    int cur = 0;
    for (int it = 0; it < iters; ++it) {
        long long nidx = idx + stride;
        if (it + 1 < iters) {
            // Kick off next tile into the other buffer, then wait only for
            // the *older* transfer (ASYNCcnt <= 1): true double buffering.
            unsigned ldsn = (cur == 0) ? lds1 : lds0;
            unsigned long long g = (unsigned long long)(uintptr_t)(in + nidx);
            asm volatile("global_load_async_to_lds_b128 %0, %1, off"
                         :: "v"(ldsn), "v"(g) : "memory");
            asm volatile("s_wait_asynccnt 0x1" ::: "memory");
        } else {
            asm volatile("s_wait_asynccnt 0x0" ::: "memory");
        }

        // Consume current tile from LDS (ds_load_b128), apply select in VALU.
        v4f v = smem[cur][tid];
        v.x = sel_one(v.x);
        v.y = sel_one(v.y);
        v.z = sel_one(v.z);
        v.w = sel_one(v.w);

        // Streaming store: nothing is re-read, keep it out of the caches.
        __builtin_nontemporal_store(v, out + idx);

        idx = nidx;
        cur ^= 1;
    }
}

// Generic scalar tail (zero-sized for the 2^27-element reference shape).
__global__ void copy_select_tail(const float* __restrict__ in,
                                 float* __restrict__ out,
                                 long long start, long long n)
{
    long long stride = (long long)gridDim.x * blockDim.x;
    for (long long i = start + (long long)blockIdx.x * blockDim.x + threadIdx.x;
         i < n; i += stride) {
        out[i] = sel_one(in[i]);
    }
}

extern "C" void kernel_launch(void* const* d_in, const int* in_sizes, int n_in,
                              void* d_out, int out_size, void* d_ws, size_t ws_size,
                              hipStream_t stream) {
    (void)n_in; (void)d_ws; (void)ws_size;

    const float* in = (const float*)d_in[0];
    float* out = (float*)d_out;
    long long n = (long long)in_sizes[0];   // == out_size

    long long n4 = n / 4;                               // # of float4 elements
    long long total = (long long)GRID * BLOCK;          // threads in bulk grid
    int iters = (int)(n4 / total);                      // uniform per-thread work
    long long covered = (long long)iters * total * 4;   // scalars handled by bulk

    if (iters > 0) {
        copy_select_async<<<GRID, BLOCK, 0, stream>>>(
            (const v4f*)in, (v4f*)out, iters, total);
    }

    long long tail = n - covered;
    if (tail > 0) {
        int tb = 256;
        long long tg_ll = (tail + tb - 1) / tb;
        int tg = (tg_ll > 4096) ? 4096 : (int)tg_ll;
        copy_select_tail<<<tg, tb, 0, stream>>>(in, out, covered, n);
    }
}